// Resonance_50861002719920
// MI455X (gfx1250) — compile-verified
//
#include <hip/hip_runtime.h>
#include <hip/hip_bf16.h>

// ---------------------------------------------------------------------------
// Problem constants (from reference)
// ---------------------------------------------------------------------------
#define N_SAMPLES   32768
#define FILTER_SIZE 512
#define N_FILTERS   128
#define MODEL_DIM   128
#define BATCH       4
#define N_EVENTS    8
#define BE          (BATCH * N_EVENTS)   // 32

typedef float v2f __attribute__((ext_vector_type(2)));
typedef float v8f __attribute__((ext_vector_type(8)));
typedef float v4f __attribute__((ext_vector_type(4)));

// ---------------------------------------------------------------------------
// CDNA5 async global->LDS staging (ASYNCcnt-tracked, no VGPR round trip).
// Probe result: __builtin_amdgcn_global_load_async_to_lds_b128 takes
//   (int4 addrspace(1)*, int4 addrspace(3)*, imm offset, imm cpol)
// where int4 is a GCC-style 16-byte int vector and AS1 prints as __device__.
// ---------------------------------------------------------------------------
typedef int v4i_gcc __attribute__((vector_size(16)));
typedef __attribute__((address_space(1))) v4i_gcc* gptr_b128;
typedef __attribute__((address_space(3))) v4i_gcc* lptr_b128;

#if defined(__has_builtin)
#  if __has_builtin(__builtin_amdgcn_global_load_async_to_lds_b128)
#    define HAVE_ASYNC_B128 1
#  endif
#endif

__device__ __forceinline__ void async_copy_b128(const float* gsrc, float* ldst) {
#if defined(HAVE_ASYNC_B128)
    __builtin_amdgcn_global_load_async_to_lds_b128(
        (gptr_b128)(void*)(const_cast<float*>(gsrc)),
        (lptr_b128)(void*)ldst, 0, 0);
#else
    *(v4f*)ldst = *(const v4f*)gsrc;
#endif
}

__device__ __forceinline__ void wait_async_then_barrier() {
#if defined(HAVE_ASYNC_B128)
#  if __has_builtin(__builtin_amdgcn_s_wait_asynccnt)
    __builtin_amdgcn_s_wait_asynccnt(0);
#  else
    asm volatile("s_wait_asynccnt 0x0" ::: "memory");
#  endif
#endif
    __syncthreads();
}

// ---------------------------------------------------------------------------
// Kernel 1: MLP (3 layers, WMMA f32 16x16x4) + combined filter g = mx @ filters
// One block, 256 threads = 8 waves. Tiles 16x16; K stepped by 4 per WMMA.
//
// WMMA f32 16x16x4 operand layout (CDNA5 ISA 7.12.2):
//   A (16x4):  lane L in [0,16): {A[L,0], A[L,1]}; lane 16+L: {A[L,2], A[L,3]}
//   B (4x16):  lane L in [0,16): col L, {B[0,L], B[1,L]}; lane 16+L: {B[2,L], B[3,L]}
//   C/D:       VGPR q: lanes 0-15 -> row q, lanes 16-31 -> row 8+q; col = lane&15
// ---------------------------------------------------------------------------
__device__ __forceinline__ void mlp_layer_16x16(
    const float (*__restrict__ hin)[MODEL_DIM],
    float (*__restrict__ hout)[MODEL_DIM],
    const float* __restrict__ W,   // (128, 128) row-major: W[k*128 + n]
    const float* __restrict__ Bb,  // (128,)
    bool square, int wave, int row, int khalf)
{
    #pragma unroll
    for (int r = 0; r < 2; ++r) {
        const int tileid = wave * 2 + r;      // 0..15
        const int mi  = tileid >> 3;          // 0..1
        const int nj  = tileid & 7;           // 0..7
        const int col = nj * 16 + row;

        v8f acc;
        const float bv = Bb[col];
        #pragma unroll
        for (int q = 0; q < 8; ++q) acc[q] = bv;

        #pragma unroll
        for (int k0 = 0; k0 < MODEL_DIM; k0 += 4) {
            const int kk = k0 + khalf * 2;
            v2f a, b;
            a.x = hin[mi * 16 + row][kk];
            a.y = hin[mi * 16 + row][kk + 1];
            b.x = W[(kk)     * MODEL_DIM + col];
            b.y = W[(kk + 1) * MODEL_DIM + col];
            acc = __builtin_amdgcn_wmma_f32_16x16x4_f32(
                false, a, false, b, (short)0, acc, false, false);
        }

        #pragma unroll
        for (int q = 0; q < 8; ++q) {
            const int m = mi * 16 + khalf * 8 + q;
            float v = acc[q];
            v = square ? (v * v) : fmaxf(v, 0.2f * v);  // leaky_relu(0.2) or (.)^2
            hout[m][col] = v;
        }
    }
    __syncthreads();
}

__global__ void __launch_bounds__(256)
resonance_mlp_filters_kernel(
    const float* __restrict__ x,        // (32, 128)
    const float* __restrict__ filters,  // (128, 512)
    const float* __restrict__ w1, const float* __restrict__ b1,
    const float* __restrict__ w2, const float* __restrict__ b2,
    const float* __restrict__ w3, const float* __restrict__ b3,
    float* __restrict__ g)              // (32, 512) workspace
{
    __shared__ float hA[BE][MODEL_DIM];
    __shared__ float hB[BE][MODEL_DIM];

    const int tid   = threadIdx.x;
    const int lane  = tid & 31;
    const int wave  = tid >> 5;     // 0..7
    const int row   = lane & 15;
    const int khalf = lane >> 4;

    // Stage x -> LDS (async b128)
    for (int v = tid; v < (BE * MODEL_DIM) / 4; v += 256)
        async_copy_b128(x + v * 4, &hA[0][0] + v * 4);
    wait_async_then_barrier();

    // h1 = lrelu(x@w1+b1); h2 = lrelu(h1@w2+b2); mx = (h2@w3+b3)^2
    mlp_layer_16x16(hA, hB, w1, b1, /*square=*/false, wave, row, khalf);
    mlp_layer_16x16(hB, hA, w2, b2, /*square=*/false, wave, row, khalf);
    mlp_layer_16x16(hA, hB, w3, b3, /*square=*/true,  wave, row, khalf);
    // mx now lives in hB (32 x 128)

    // g = mx @ filters : M=32 (2 tiles), N=512 (32 tiles), K=128 -> 64 tiles, 8/wave
    #pragma unroll
    for (int r = 0; r < 8; ++r) {
        const int tileid = wave * 8 + r;   // 0..63
        const int mi  = tileid >> 5;
        const int nj  = tileid & 31;
        const int col = nj * 16 + row;

        v8f acc = {};
        #pragma unroll
        for (int k0 = 0; k0 < MODEL_DIM; k0 += 4) {
            const int kk = k0 + khalf * 2;
            v2f a, b;
            a.x = hB[mi * 16 + row][kk];
            a.y = hB[mi * 16 + row][kk + 1];
            b.x = filters[(kk)     * FILTER_SIZE + col];
            b.y = filters[(kk + 1) * FILTER_SIZE + col];
            acc = __builtin_amdgcn_wmma_f32_16x16x4_f32(
                false, a, false, b, (short)0, acc, false, false);
        }
        #pragma unroll
        for (int q = 0; q < 8; ++q) {
            const int m = mi * 16 + khalf * 8 + q;
            g[m * FILTER_SIZE + col] = acc[q];
        }
    }
}

// ---------------------------------------------------------------------------
// Kernel 2: collapsed FIR.
//   out[be, s] = sum_{t=0}^{511} g[be,t] * sig[be, s - 256 + t]   (zero-padded)
// Grid: (8 chunks of 4096 samples, 32 be). 256 threads.
// Signal chunk + 512-sample halo and g[be] staged into LDS with
// GLOBAL_LOAD_ASYNC_TO_LDS_B128 (interior) / predicated VGPR copies (edges).
// Each thread: 16 outputs = 4 groups of 4 consecutive (16B aligned).
// Inner step: 1 broadcast b128 (g), 8 aligned b128 (sig window), 64 FMAs.
// Results stored nontemporal (streamed, never re-read).
// ---------------------------------------------------------------------------
#define CHUNK 4096
#define SSTAGE (CHUNK + FILTER_SIZE)   // 4608 floats

__global__ void __launch_bounds__(256)
resonance_fir_kernel(const float* __restrict__ signal,  // (32, 32768)
                     const float* __restrict__ g,       // (32, 512)
                     float* __restrict__ out)           // (32, 32768)
{
    __shared__ float sigS[SSTAGE + 8];
    __shared__ float gS[FILTER_SIZE];

    const int be  = blockIdx.y;
    const int s0  = blockIdx.x * CHUNK;
    const int tid = threadIdx.x;
    const float* sig = signal + be * N_SAMPLES;

    // Stage combined filter: 128 x b128, always in range
    for (int v = tid; v < FILTER_SIZE / 4; v += 256)
        async_copy_b128(g + be * FILTER_SIZE + v * 4, &gS[v * 4]);

    // Stage signal chunk + halo; zero-pad outside [0, N_SAMPLES)
    for (int v = tid; v < SSTAGE / 4; v += 256) {
        const int i   = v * 4;
        const int idx = s0 - (FILTER_SIZE / 2) + i;
        if (idx >= 0 && idx + 3 < N_SAMPLES) {
            async_copy_b128(sig + idx, &sigS[i]);
        } else {
            #pragma unroll
            for (int j = 0; j < 4; ++j) {
                const int ij = idx + j;
                sigS[i + j] = (ij >= 0 && ij < N_SAMPLES) ? sig[ij] : 0.0f;
            }
        }
    }
    wait_async_then_barrier();

    float acc[4][4] = {};
    #pragma unroll 2
    for (int t = 0; t < FILTER_SIZE; t += 4) {
        const v4f gv = *(const v4f*)&gS[t];
        #pragma unroll
        for (int rep = 0; rep < 4; ++rep) {
            const int off = rep * 1024 + tid * 4;       // output group base
            const v4f wa = *(const v4f*)&sigS[off + t];
            const v4f wb = *(const v4f*)&sigS[off + t + 4];
            const float w[8] = {wa.x, wa.y, wa.z, wa.w, wb.x, wb.y, wb.z, wb.w};
            #pragma unroll
            for (int j = 0; j < 4; ++j) {
                float a = acc[rep][j];
                a = fmaf(gv.x, w[j],     a);
                a = fmaf(gv.y, w[j + 1], a);
                a = fmaf(gv.z, w[j + 2], a);
                a = fmaf(gv.w, w[j + 3], a);
                acc[rep][j] = a;
            }
        }
    }

    float* o = out + be * N_SAMPLES + s0;
    #pragma unroll
    for (int rep = 0; rep < 4; ++rep) {
        const int off = rep * 1024 + tid * 4;
        v4f v = {acc[rep][0], acc[rep][1], acc[rep][2], acc[rep][3]};
        __builtin_nontemporal_store(v, (v4f*)&o[off]);
    }
}

// ---------------------------------------------------------------------------
// Launch
// Inputs (setup_inputs order): signal, x, filters, w1, b1, w2, b2, w3, b3
// Output: (4, 8, 1, 32768) f32  == (32, 32768)
// Workspace: g (32*512 f32 = 64 KB)
// ---------------------------------------------------------------------------
extern "C" void kernel_launch(void* const* d_in, const int* in_sizes, int n_in,
                              void* d_out, int out_size, void* d_ws, size_t ws_size,
                              hipStream_t stream) {
    const float* signal  = (const float*)d_in[0];
    const float* x       = (const float*)d_in[1];
    const float* filters = (const float*)d_in[2];
    const float* w1      = (const float*)d_in[3];
    const float* b1      = (const float*)d_in[4];
    const float* w2      = (const float*)d_in[5];
    const float* b2      = (const float*)d_in[6];
    const float* w3      = (const float*)d_in[7];
    const float* b3      = (const float*)d_in[8];

    float* g   = (float*)d_ws;          // 32 * 512 floats
    float* out = (float*)d_out;         // 32 * 32768 floats

    resonance_mlp_filters_kernel<<<1, 256, 0, stream>>>(
        x, filters, w1, b1, w2, b2, w3, b3, g);

    dim3 grid(N_SAMPLES / CHUNK, BE);   // (8, 32)
    resonance_fir_kernel<<<grid, 256, 0, stream>>>(signal, g, out);
}